// PhysicsInformedODE_39754217292300
// MI455X (gfx1250) — compile-verified
//
// CDNA5 / gfx1250 — physics-informed ODE RK4 integrator.
// Design: one trajectory per lane, 32 per wave, single-wave workgroups.
// All five output GEMM chains (fourier-linear, quad MLP L2, cubic MLP L2,
// T-net L2, H-net L2) accumulate into a single v_wmma_f32_16x16x4_f32
// C accumulator (out^T, M=state dims, N=batch lanes). Layer-1 biases are
// folded in as a constant-1 K-feature. C->B layout conversion between
// chained WMMAs uses ds_swizzle SWAPX16 + half-lane selects.
#include <hip/hip_runtime.h>

typedef __attribute__((ext_vector_type(2))) float v2f;
typedef __attribute__((ext_vector_type(8))) float v8f;

#define SROW 34                 // staging row stride (dwords): even (b64 align), conflict-free
#define TWO_PI 6.283185307179586f

// --- lane-half swap via LDS swizzle (group-of-32: xor=0x10, and=0x1f) ---
__device__ __forceinline__ float swap16(float x) {
  return __int_as_float(__builtin_amdgcn_ds_swizzle(__float_as_int(x), 0x401f));
}
__device__ __forceinline__ float sel16(float lo, float hi, int lane) {
  return (lane & 16) ? hi : lo;
}
// --- hardware tanh (TRANS op: keep a v_nop after per ISA hazard rule) ---
__device__ __forceinline__ float vtanh(float x) {
  float r;
  asm volatile("v_tanh_f32 %0, %1\n\tv_nop" : "=v"(r) : "v"(x));
  return r;
}
// --- f32 WMMA 16x16x4: D = A(16x4) * B(4x16) + C ---
__device__ __forceinline__ v8f wmma4(v2f a, v2f b, v8f c) {
  return __builtin_amdgcn_wmma_f32_16x16x4_f32(false, a, false, b, (short)0, c,
                                               false, false);
}
// B-operand load from staged LDS row: reg j holds K = 4*ks + j + kh
__device__ __forceinline__ v2f bld(const float* Brow, int off, int ks, int kh) {
  float2 t = *(const float2*)(Brow + off + 4 * ks + kh);
  v2f b; b.x = t.x; b.y = t.y; return b;
}
// Build B-layout reg pair (K rows 4kk..4kk+3) from a C-layout array
// (C reg v pairs rows v / v+8) using half-swaps.
__device__ __forceinline__ v2f bfromc(const float c[8], int kk, int lane) {
  v2f b;
  if (kk < 2) {
    b.x = sel16(c[4 * kk + 0], swap16(c[4 * kk + 2]), lane);
    b.y = sel16(c[4 * kk + 1], swap16(c[4 * kk + 3]), lane);
  } else {
    int q = kk - 2;
    b.x = sel16(swap16(c[4 * q + 0]), c[4 * q + 2], lane);
    b.y = sel16(swap16(c[4 * q + 1]), c[4 * q + 3], lane);
  }
  return b;
}
// Layer-1: h^T = tanh(W1 * feats^T) — bias folded into K (constant-1 feature)
__device__ __forceinline__ void l1chain(const v2f A[2], const v2f b[2],
                                        float hout[8]) {
  v8f c = {0.f, 0.f, 0.f, 0.f, 0.f, 0.f, 0.f, 0.f};
  c = wmma4(A[0], b[0], c);
  c = wmma4(A[1], b[1], c);
#pragma unroll
  for (int v = 0; v < 8; ++v) hout[v] = vtanh(c[v]);
}

__global__ __launch_bounds__(32) void ode_rk4_wmma(
    const float* __restrict__ x0, const float* __restrict__ tgrid,
    const float* __restrict__ fourier,
    const float* __restrict__ qW1, const float* __restrict__ qb1,
    const float* __restrict__ qW2, const float* __restrict__ qb2,
    const float* __restrict__ cW1, const float* __restrict__ cb1,
    const float* __restrict__ cW2, const float* __restrict__ cb2,
    const float* __restrict__ tW1, const float* __restrict__ tb1,
    const float* __restrict__ tW2, const float* __restrict__ tb2,
    const float* __restrict__ hW1, const float* __restrict__ hb1,
    const float* __restrict__ hW2, const float* __restrict__ hb2,
    float* __restrict__ out, int B, int T) {
  __shared__ float Fsm[180];        // fourier (6,6,5)
  __shared__ float Lsm[108];        // L(t0), L(t0+h), L(t0+dt), row-major 6x6
  __shared__ float Ssm[32 * SROW];  // per-lane staged features

  const int lane = threadIdx.x;
  const int r = lane & 15;                 // A/B row-within-tile index
  const int kh = (lane & 16) ? 2 : 0;      // K offset for upper lane half
  const int b = blockIdx.x * 32 + lane;
  const int bl = (b < B) ? b : (B - 1);    // clamp: keep EXEC full for WMMA

  for (int e = lane; e < 180; e += 32) Fsm[e] = fourier[e];

  {  // zero staging row; plant constant-1 bias features (x2@14, x3@22, feats@29)
    float* Srow = &Ssm[lane * SROW];
#pragma unroll
    for (int s2 = 0; s2 < SROW; ++s2) Srow[s2] = 0.f;
    Srow[14] = 1.f; Srow[22] = 1.f; Srow[29] = 1.f;
  }

  // ---- constant weights pre-swizzled into WMMA A-layout registers ----
  // A layout (f32 16x4): lanes 0-15 hold (M=r, K=4ks+j), lanes 16-31 (M=r, K=4ks+2+j)
  v2f Aq1[2][2], Ac1[2][2], At1[2], Ah1[2];
#pragma unroll
  for (int m = 0; m < 2; ++m)
#pragma unroll
    for (int ks = 0; ks < 2; ++ks)
#pragma unroll
      for (int j = 0; j < 2; ++j) {
        int col = 4 * ks + j + kh;
        int row = 16 * m + r;
        float aq = 0.f, ac = 0.f;
        if (col < 6)       { aq = qW1[row * 6 + col]; ac = cW1[row * 6 + col]; }
        else if (col == 6) { aq = qb1[row];           ac = cb1[row]; }   // bias-as-K
        Aq1[m][ks][j] = aq; Ac1[m][ks][j] = ac;
      }
#pragma unroll
  for (int ks = 0; ks < 2; ++ks)
#pragma unroll
    for (int j = 0; j < 2; ++j) {
      int col = 4 * ks + j + kh;
      float at = 0.f, ah = 0.f;
      if (col < 5)       { at = tW1[r * 5 + col]; ah = hW1[r * 5 + col]; }
      else if (col == 5) { at = tb1[r];           ah = hb1[r]; }
      At1[ks][j] = at; Ah1[ks][j] = ah;
    }
  v2f Aq2[8], Ac2[8];
#pragma unroll
  for (int ks = 0; ks < 8; ++ks)
#pragma unroll
    for (int j = 0; j < 2; ++j) {
      int col = 4 * ks + j + kh;
      Aq2[ks][j] = (r < 6) ? qW2[r * 32 + col] : 0.f;
      Ac2[ks][j] = (r < 6) ? cW2[r * 32 + col] : 0.f;
    }
  v2f At2[4], Ah2[4];  // 1x16 row vectors placed on A rows 0 (eT) and 1 (eH)
#pragma unroll
  for (int ks = 0; ks < 4; ++ks)
#pragma unroll
    for (int j = 0; j < 2; ++j) {
      int col = 4 * ks + j + kh;
      At2[ks][j] = (r == 0) ? tW2[col] : 0.f;
      Ah2[ks][j] = (r == 1) ? hW2[col] : 0.f;
    }
  float bias2[6];
#pragma unroll
  for (int d = 0; d < 6; ++d) bias2[d] = qb2[d] + cb2[d];
  bias2[0] += tb2[0];
  bias2[1] += hb2[0];

  float xcur[6];
#pragma unroll
  for (int d = 0; d < 6; ++d) xcur[d] = x0[bl * 6 + d];
  if (b < B) {
    float2* po = (float2*)&out[(size_t)b * T * 6];
    po[0] = make_float2(xcur[0], xcur[1]);
    po[1] = make_float2(xcur[2], xcur[3]);
    po[2] = make_float2(xcur[4], xcur[5]);
  }
  __syncthreads();  // single-wave WG: barrier is S_NOP, keeps LDS ordering

  // ---- one RHS evaluation: k = L(ts)x + MLPs(x) ----
  auto rhs = [&](const float (&xin)[6], int Lm, float (&kout)[6]) {
    float x2v[6], x3v[6];
#pragma unroll
    for (int d = 0; d < 6; ++d) { x2v[d] = xin[d] * xin[d]; x3v[d] = x2v[d] * xin[d]; }
    float Tv = xin[0], Hv = xin[1];
    float f0 = Tv * Tv, f1 = Tv * Hv, f2 = f0 * Tv, f3 = f0 * Hv, f4 = f1 * Hv;
    float* Srow = &Ssm[lane * SROW];
    ((float2*)Srow)[0] = make_float2(xin[0], xin[1]);
    ((float2*)Srow)[1] = make_float2(xin[2], xin[3]);
    ((float2*)Srow)[2] = make_float2(xin[4], xin[5]);
    ((float2*)(Srow + 8))[0]  = make_float2(x2v[0], x2v[1]);
    ((float2*)(Srow + 8))[1]  = make_float2(x2v[2], x2v[3]);
    ((float2*)(Srow + 8))[2]  = make_float2(x2v[4], x2v[5]);
    ((float2*)(Srow + 16))[0] = make_float2(x3v[0], x3v[1]);
    ((float2*)(Srow + 16))[1] = make_float2(x3v[2], x3v[3]);
    ((float2*)(Srow + 16))[2] = make_float2(x3v[4], x3v[5]);
    ((float2*)(Srow + 24))[0] = make_float2(f0, f1);
    ((float2*)(Srow + 24))[1] = make_float2(f2, f3);
    Srow[28] = f4;
    __syncthreads();

    v2f AL[2];  // time-dependent 6x6 L matrix into A layout (per RHS)
#pragma unroll
    for (int ks = 0; ks < 2; ++ks)
#pragma unroll
      for (int j = 0; j < 2; ++j) {
        int col = 4 * ks + j + kh;
        bool ok = (r < 6) && (col < 6);
        float val = Lsm[Lm * 36 + (ok ? (r * 6 + col) : 0)];
        AL[ks][j] = ok ? val : 0.f;
      }

    v8f caccs[2];
#pragma unroll
    for (int n = 0; n < 2; ++n) {  // two 16-batch N-tiles
      const float* Brow = &Ssm[(n * 16 + r) * SROW];
      v2f bx[2], b2[2], b3[2], bf[2];
#pragma unroll
      for (int ks = 0; ks < 2; ++ks) {
        bx[ks] = bld(Brow, 0, ks, kh);
        b2[ks] = bld(Brow, 8, ks, kh);
        b3[ks] = bld(Brow, 16, ks, kh);
        bf[ks] = bld(Brow, 24, ks, kh);
      }
      float hq0[8], hq1[8], hc0[8], hc1[8], ht[8], hh[8];
      l1chain(Aq1[0], b2, hq0);   // quad hidden 0-15
      l1chain(Aq1[1], b2, hq1);   // quad hidden 16-31
      l1chain(Ac1[0], b3, hc0);   // cubic hidden 0-15
      l1chain(Ac1[1], b3, hc1);   // cubic hidden 16-31
      l1chain(At1, bf, ht);       // T-net hidden 0-15
      l1chain(Ah1, bf, hh);       // H-net hidden 0-15

      v8f ca;  // out^T accumulator, init with layer-2 biases on rows 0..5
#pragma unroll
      for (int v = 0; v < 8; ++v) ca[v] = 0.f;
#pragma unroll
      for (int d = 0; d < 6; ++d) ca[d] = (lane < 16) ? bias2[d] : 0.f;

      ca = wmma4(AL[0], bx[0], ca);  // fourier-linear term
      ca = wmma4(AL[1], bx[1], ca);
#pragma unroll
      for (int kk = 0; kk < 4; ++kk) ca = wmma4(Aq2[kk],     bfromc(hq0, kk, lane), ca);
#pragma unroll
      for (int kk = 0; kk < 4; ++kk) ca = wmma4(Aq2[kk + 4], bfromc(hq1, kk, lane), ca);
#pragma unroll
      for (int kk = 0; kk < 4; ++kk) ca = wmma4(Ac2[kk],     bfromc(hc0, kk, lane), ca);
#pragma unroll
      for (int kk = 0; kk < 4; ++kk) ca = wmma4(Ac2[kk + 4], bfromc(hc1, kk, lane), ca);
#pragma unroll
      for (int kk = 0; kk < 4; ++kk) ca = wmma4(At2[kk],     bfromc(ht, kk, lane), ca);
#pragma unroll
      for (int kk = 0; kk < 4; ++kk) ca = wmma4(Ah2[kk],     bfromc(hh, kk, lane), ca);
      caccs[n] = ca;
    }
#pragma unroll
    for (int d = 0; d < 6; ++d)  // un-stage: lane L gets out[d] of batch L
      kout[d] = sel16(caccs[0][d], swap16(caccs[1][d]), lane);
    __syncthreads();
  };

  for (int i = 0; i < T - 1; ++i) {
    float t0 = tgrid[i];
    float dt = tgrid[i + 1] - t0;
    float hf = 0.5f * dt;
    {  // cooperative build of L(t0), L(t0+h), L(t0+dt) (36 elems each)
      for (int e = lane; e < 108; e += 32) {
        int m = (e >= 72) ? 2 : (e >= 36 ? 1 : 0);
        int idx = e - m * 36;
        float ts = (m == 0) ? t0 : ((m == 1) ? (t0 + hf) : (t0 + dt));
        float ang = TWO_PI * ts;
        float s1 = __sinf(ang), c1 = __cosf(ang);
        float c2 = c1 * c1 - s1 * s1, s2 = 2.f * s1 * c1;
        const float* f = &Fsm[idx * 5];
        Lsm[e] = f[0] + c1 * f[1] + c2 * f[2] + s1 * f[3] + s2 * f[4];
      }
    }
    float k1[6], k2[6], k3[6], k4[6], xs[6];
    rhs(xcur, 0, k1);
#pragma unroll
    for (int d = 0; d < 6; ++d) xs[d] = xcur[d] + hf * k1[d];
    rhs(xs, 1, k2);
#pragma unroll
    for (int d = 0; d < 6; ++d) xs[d] = xcur[d] + hf * k2[d];
    rhs(xs, 1, k3);
#pragma unroll
    for (int d = 0; d < 6; ++d) xs[d] = xcur[d] + dt * k3[d];
    rhs(xs, 2, k4);
    float sixth = dt * (1.f / 6.f);
#pragma unroll
    for (int d = 0; d < 6; ++d)
      xcur[d] += sixth * (k1[d] + 2.f * (k2[d] + k3[d]) + k4[d]);
    if (b < B) {
      float2* po = (float2*)&out[((size_t)b * T + (i + 1)) * 6];
      po[0] = make_float2(xcur[0], xcur[1]);
      po[1] = make_float2(xcur[2], xcur[3]);
      po[2] = make_float2(xcur[4], xcur[5]);
    }
  }
}

extern "C" void kernel_launch(void* const* d_in, const int* in_sizes, int n_in,
                              void* d_out, int out_size, void* d_ws,
                              size_t ws_size, hipStream_t stream) {
  (void)n_in; (void)d_ws; (void)ws_size; (void)out_size;
  const int B = in_sizes[0] / 6;
  const int T = in_sizes[1];
  dim3 grid((B + 31) / 32), block(32);
  hipLaunchKernelGGL(ode_rk4_wmma, grid, block, 0, stream,
                     (const float*)d_in[0],  (const float*)d_in[1],
                     (const float*)d_in[2],
                     (const float*)d_in[3],  (const float*)d_in[4],
                     (const float*)d_in[5],  (const float*)d_in[6],
                     (const float*)d_in[7],  (const float*)d_in[8],
                     (const float*)d_in[9],  (const float*)d_in[10],
                     (const float*)d_in[11], (const float*)d_in[12],
                     (const float*)d_in[13], (const float*)d_in[14],
                     (const float*)d_in[15], (const float*)d_in[16],
                     (const float*)d_in[17], (const float*)d_in[18],
                     (float*)d_out, B, T);
}